// SequentialSparsemax_10720238370935
// MI455X (gfx1250) — compile-verified
//
#include <hip/hip_runtime.h>

// Fused sequential-sparsemax for (B=8, N=16, T=262144) f32, LST=64.
// Memory-bound: 256 MiB traffic -> ~11 us roofline at 23.3 TB/s.
// Pipeline per 16x64 tile (one frame, all instruments):
//   TDM tensor_load_to_lds (2D strided tile, pad_enable -> row pitch 65 dwords)
//   -> Michelot sparsemax over instruments (d=16, thread-per-column)
//   -> Michelot sparsemax over the frame (d=64, thread-per-row)
//   -> coalesced full-cacheline nontemporal global_store_b128.

#define T_LEN   262144
#define N_INST  16
#define LST     64
#define FRAMES  (T_LEN / LST)          // 4096 frames per (b, inst)
#define B_DIM   8
#define TILES_PER_BLOCK 8
#define BLOCK   128                    // 4 waves (wave32)
#define ROW_PITCH 65                   // padded dwords per LDS row (bank-conflict free)
#define TILE_DW  (N_INST * ROW_PITCH)  // 1040 dwords per tile
#define LDS_DW   (TILES_PER_BLOCK * TILE_DW)  // 8320 dwords = 33280 bytes

#ifndef __has_builtin
#define __has_builtin(x) 0
#endif

#if __has_builtin(__builtin_amdgcn_tensor_load_to_lds)
#define USE_TDM 1
#else
#define USE_TDM 0
#endif

typedef __attribute__((ext_vector_type(4))) unsigned int u32x4;
typedef __attribute__((ext_vector_type(8))) int          i32x8;
typedef __attribute__((ext_vector_type(4))) int          i32x4;
typedef __attribute__((ext_vector_type(4))) float        f32x4;

__device__ __forceinline__ void wait_tensorcnt0() {
#if __has_builtin(__builtin_amdgcn_s_wait_tensorcnt)
    __builtin_amdgcn_s_wait_tensorcnt(0);
#else
    asm volatile("s_wait_tensorcnt 0x0" ::: "memory");
#endif
}

__global__ __launch_bounds__(BLOCK)
void fused_seq_sparsemax_kernel(const float* __restrict__ in, float* __restrict__ out)
{
    __shared__ float lds[LDS_DW];
    const int tid = threadIdx.x;
    const int bid = blockIdx.x;

#if USE_TDM
    // ---- Stage tiles global -> LDS with the Tensor Data Mover -------------
    // Each wave issues 2 tile descriptors (8 tiles/block). 2D tile:
    // tile_dim0=64 elems (a frame), tile_dim1=16 rows (instruments),
    // tensor_dim0_stride = T_LEN, data_size=4B. pad_enable inserts 1 dword
    // after every 64 dwords -> LDS row pitch of 65 dwords.
    {
        const int wave = tid >> 5;
        #pragma unroll
        for (int q = 0; q < 2; ++q) {
            const int lt = wave * 2 + q;
            const int g  = bid * TILES_PER_BLOCK + lt;     // global tile id
            const int b  = g >> 12;                        // / FRAMES
            const int f  = g & (FRAMES - 1);
            const unsigned long long gaddr =
                (unsigned long long)(uintptr_t)in +
                (((unsigned long long)b * N_INST * T_LEN) +
                 ((unsigned long long)f * LST)) * 4ull;
            // Flat LDS pointer: low 32 bits are the LDS byte address.
            const unsigned lds_byte =
                (unsigned)(uintptr_t)(&lds[lt * TILE_DW]);

            u32x4 g0; i32x8 g1;
            i32x4 g2 = {0, 0, 0, 0};
            i32x4 g3 = {0, 0, 0, 0};
            // Group 0: count=1 (valid user D#), lds_addr, global_addr, type=2
            g0[0] = 1u;
            g0[1] = lds_byte;
            g0[2] = (unsigned)(gaddr & 0xffffffffull);
            g0[3] = ((unsigned)(gaddr >> 32) & 0x01ffffffu) | (2u << 30);
            // Group 1
            g1[0] = (2 << 16)      // data_size = 4 bytes
                  | (1 << 20)      // pad_enable
                  | (5 << 22);     // pad_interval: every 64 dwords (pad_amount=0 -> 1 dword)
            g1[1] = 0;                                        // tensor_dim0 lo16 = 0
            g1[2] = (int)((T_LEN >> 16) | (N_INST << 16));    // tensor_dim0 hi / tensor_dim1 lo
            g1[3] = (int)(LST << 16);                         // tile_dim0 = 64
            g1[4] = N_INST;                                   // tile_dim1 = 16, tile_dim2 = 0
            g1[5] = T_LEN;                                    // tensor_dim0_stride lo32
            g1[6] = 0;
            g1[7] = 0;
#if defined(__clang_major__) && __clang_major__ >= 23
            i32x8 g2b = {0, 0, 0, 0, 0, 0, 0, 0};
            __builtin_amdgcn_tensor_load_to_lds(g0, g1, g2, g3, g2b, 0);
#else
            __builtin_amdgcn_tensor_load_to_lds(g0, g1, g2, g3, 0);
#endif
        }
        wait_tensorcnt0();
    }
#else
    // Fallback: cooperative copy global -> padded LDS tiles.
    for (int i = tid; i < TILES_PER_BLOCK * N_INST * LST; i += BLOCK) {
        const int lt  = i >> 10;
        const int rem = i & 1023;
        const int r   = rem >> 6;
        const int c   = rem & 63;
        const int g   = bid * TILES_PER_BLOCK + lt;
        const int b   = g >> 12;
        const int f   = g & (FRAMES - 1);
        lds[lt * TILE_DW + r * ROW_PITCH + c] =
            in[(size_t)(b * N_INST + r) * T_LEN + (size_t)f * LST + c];
    }
#endif
    __syncthreads();

    // ---- Pass 1: sparsemax over the 16-instrument axis, per time column ---
    // 8 tiles * 64 columns = 512 columns; 128 threads -> 4 columns each.
    // LDS banks: addr = 16*lt + i + j (mod 64); j consecutive across lanes
    // -> conflict free.
    #pragma unroll
    for (int q = 0; q < 4; ++q) {
        const int c  = tid + q * BLOCK;     // 0..511
        const int lt = c >> 6;
        const int j  = c & 63;
        float* col = &lds[lt * TILE_DW + j];
        float z[N_INST];
        #pragma unroll
        for (int i = 0; i < N_INST; ++i) z[i] = col[i * ROW_PITCH];
        float sum = 0.0f;
        #pragma unroll
        for (int i = 0; i < N_INST; ++i) sum += z[i];
        int   k   = N_INST;
        float tau = (sum - 1.0f) * (1.0f / (float)N_INST);
        // Michelot projection onto the simplex: drop entries <= tau until stable.
        for (int it = 0; it < N_INST; ++it) {
            float s = 0.0f; int cnt = 0;
            #pragma unroll
            for (int i = 0; i < N_INST; ++i) {
                const bool inq = z[i] > tau;
                s   = inq ? (s + z[i]) : s;
                cnt += inq ? 1 : 0;
            }
            if (cnt == k || cnt == 0) break;
            k = cnt; tau = (s - 1.0f) / (float)cnt;
        }
        #pragma unroll
        for (int i = 0; i < N_INST; ++i)
            col[i * ROW_PITCH] = fmaxf(z[i] - tau, 0.0f);
    }
    __syncthreads();

    // ---- Pass 2: sparsemax over each 64-sample frame, per instrument row --
    // 8 tiles * 16 rows = 128 rows; one thread per row. Banks: 16*lt + r + i
    // (mod 64) distinct across lanes -> conflict-free b32 loads/stores.
    {
        const int lt = tid >> 4;
        const int r  = tid & 15;
        float* row = &lds[lt * TILE_DW + r * ROW_PITCH];
        float z[LST];
        #pragma unroll
        for (int i = 0; i < LST; ++i) z[i] = row[i];
        float sum = 0.0f;
        #pragma unroll
        for (int i = 0; i < LST; ++i) sum += z[i];
        int   k   = LST;
        float tau = (sum - 1.0f) * (1.0f / (float)LST);
        for (int it = 0; it < LST; ++it) {
            float s = 0.0f; int cnt = 0;
            #pragma unroll
            for (int i = 0; i < LST; ++i) {
                const bool inq = z[i] > tau;
                s   = inq ? (s + z[i]) : s;
                cnt += inq ? 1 : 0;
            }
            if (cnt == k || cnt == 0) break;
            k = cnt; tau = (s - 1.0f) / (float)cnt;
        }
        // Write results back to the padded LDS row (conflict-free) so the
        // store phase can emit fully-coalesced cacheline writes.
        #pragma unroll
        for (int i = 0; i < LST; ++i) row[i] = fmaxf(z[i] - tau, 0.0f);
    }
    __syncthreads();

    // ---- Store phase: full-cacheline, nontemporal global stores -----------
    // Each wave stores 32 rows (its 2 tiles' worth); per step, lanes 0..15
    // cover one row (256B contiguous) and lanes 16..31 the next row, so each
    // global_store_b128 instruction writes 4 fully-covered 128B lines.
    // LDS reads are scalar b32 (padded rows are not 16B aligned); banks
    // C + 4*l16 + j vs C+1+4*l16 + j for the two half-waves -> conflict free.
    {
        const int wave = tid >> 5;
        const int lane = tid & 31;
        const int l16  = lane & 15;
        #pragma unroll
        for (int s = 0; s < 16; ++s) {
            const int row = wave * 32 + s * 2 + (lane >> 4);
            const int lt  = row >> 4;
            const int r   = row & 15;
            const float* src = &lds[lt * TILE_DW + r * ROW_PITCH + l16 * 4];
            f32x4 v;
            v.x = src[0];
            v.y = src[1];
            v.z = src[2];
            v.w = src[3];
            const int g = bid * TILES_PER_BLOCK + lt;
            const int b = g >> 12;
            const int f = g & (FRAMES - 1);
            f32x4* dst = (f32x4*)(out + ((size_t)(b * N_INST + r) * T_LEN
                                         + (size_t)f * LST + l16 * 4));
            __builtin_nontemporal_store(v, dst);
        }
    }
}

extern "C" void kernel_launch(void* const* d_in, const int* in_sizes, int n_in,
                              void* d_out, int out_size, void* d_ws, size_t ws_size,
                              hipStream_t stream) {
    (void)in_sizes; (void)n_in; (void)d_ws; (void)ws_size; (void)out_size;
    const float* in  = (const float*)d_in[0];
    float*       out = (float*)d_out;
    const int total_tiles = B_DIM * FRAMES;                 // 32768
    const int grid = total_tiles / TILES_PER_BLOCK;         // 4096
    fused_seq_sparsemax_kernel<<<grid, BLOCK, 0, stream>>>(in, out);
}